// Sequence_27719718928976
// MI455X (gfx1250) — compile-verified
//
#include <hip/hip_runtime.h>

typedef __attribute__((ext_vector_type(2))) float v2f;
typedef __attribute__((ext_vector_type(8))) float v8f;

#define T_LEN   8192
#define HDIM    50
#define NROWS   200            // 4*H
#define KDIM    100            // 2*H
#define NCHUNK  25             // KDIM / 4
#define NWAVES  13             // ceil(200/16)
#define NTHREADS (NWAVES * 32)
#define GPITCH  216            // gates row pitch (floats), 16B-aligned
#define VPITCH  104            // per-wave [v|h] pitch (floats)

// CDNA5 hardware tanh (v_tanh_f32). TRANS ops need 1 op gap before the
// result is consumed (ISA 7.4); the builtin path lets the compiler handle it,
// the asm fallback embeds a v_nop.
__device__ __forceinline__ float tanh_hw(float x) {
#if __has_builtin(__builtin_amdgcn_tanhf)
    return __builtin_amdgcn_tanhf(x);
#else
    float r;
    asm volatile("v_tanh_f32 %0, %1\n\tv_nop" : "=v"(r) : "v"(x));
    return r;
#endif
}
__device__ __forceinline__ float sigmoid_hw(float x) {
    // sigma(x) = 0.5*tanh(x/2) + 0.5  (exact identity)
    return __builtin_fmaf(0.5f, tanh_hw(0.5f * x), 0.5f);
}

__launch_bounds__(NTHREADS, 1)
__global__ void lstm_seq_kernel(const float* __restrict__ x,
                                const float* __restrict__ W_sig,
                                const float* __restrict__ b_sig,
                                const float* __restrict__ W_ih,
                                const float* __restrict__ b_ih,
                                const float* __restrict__ W_hh,
                                const float* __restrict__ b_hh,
                                float* __restrict__ Hbuf) {
    __shared__ __align__(16) float xs[T_LEN];              // whole input (32 KB)
    __shared__ __align__(16) float gates[2][GPITCH];       // double-buffered raw gates
    __shared__ __align__(16) float vhw[NWAVES][VPITCH];    // per-wave private [v(50)|h(50)]

    const int tid  = threadIdx.x;
    const int lane = tid & 31;
    const int wave = tid >> 5;
    const int r0   = wave * 16;
    const int m    = lane & 15;
    const int hi   = (lane >= 16) ? 2 : 0;
    const int row  = r0 + m;
    float* const vh = vhw[wave];

    // ---- stage input sequence into LDS ----
    for (int i = tid; i < T_LEN; i += NTHREADS) xs[i] = x[i];

    // ---- static A fragments: W = [W_ih | W_hh], 25 chunks of K=4 ----
    float a0[NCHUNK], a1[NCHUNK];
#pragma unroll
    for (int k = 0; k < NCHUNK; ++k) {
        const int c0 = 4 * k + hi;
        float w0 = 0.f, w1 = 0.f;
        if (row < NROWS) {
            w0 = (c0     < HDIM) ? W_ih[row * HDIM + c0]     : W_hh[row * HDIM + (c0 - HDIM)];
            w1 = (c0 + 1 < HDIM) ? W_ih[row * HDIM + c0 + 1] : W_hh[row * HDIM + (c0 + 1 - HDIM)];
        }
        a0[k] = w0; a1[k] = w1;
    }

    // ---- accumulator seed = combined bias, in D layout ----
    v8f accbias;
#pragma unroll
    for (int r = 0; r < 8; ++r) {
        const int rd = r0 + r + ((lane >= 16) ? 8 : 0);
        accbias[r] = (rd < NROWS) ? (b_ih[rd] + b_hh[rd]) : 0.f;
    }

    // ---- per-lane elementwise state (replicated across waves) ----
    // lane handles j1 = lane (always < 50) and j2 = lane + 32 (if lane < 18)
    const int j1 = lane, j2 = lane + 32;
    const bool has2 = (lane < 18);
    const float ws1 = W_sig[j1], bs1 = b_sig[j1];
    const float ws2 = has2 ? W_sig[j2] : 0.f;
    const float bs2 = has2 ? b_sig[j2] : 0.f;
    float c1 = 0.f, c2 = 0.f;

    __syncthreads();                       // xs staged

    // init wave-private [v_0 | h_{-1}=0]
    vh[HDIM + j1] = 0.f;
    vh[j1] = sigmoid_hw(__builtin_fmaf(xs[0], ws1, bs1));
    if (has2) {
        vh[HDIM + j2] = 0.f;
        vh[j2] = sigmoid_hw(__builtin_fmaf(xs[0], ws2, bs2));
    }

    for (int t = 0; t < T_LEN; ++t) {
        float* const gbuf = gates[t & 1];

        // ---------- gates = W @ [v;h] + bias : 25 chained f32 WMMAs ----------
        v8f acc = accbias;
#pragma unroll
        for (int k = 0; k < NCHUNK; ++k) {
            const v2f bv = *(const v2f*)(&vh[4 * k + hi]);
            v2f av; av[0] = a0[k]; av[1] = a1[k];
            acc = __builtin_amdgcn_wmma_f32_16x16x4_f32(
                      false, av, false, bv, (short)0, acc, false, false);
        }
        if (lane == 0 || lane == 16) {     // D column N=0: lane0->M0..7, lane16->M8..15
            float4* gp = (float4*)&gbuf[r0 + ((lane == 16) ? 8 : 0)];
            gp[0] = make_float4(acc[0], acc[1], acc[2], acc[3]);
            gp[1] = make_float4(acc[4], acc[5], acc[6], acc[7]);
        }
        __syncthreads();                   // gates_t complete (only barrier per step)

        // ---------- elementwise update, replicated per wave (no 2nd barrier) ----------
        {
            const float i_ = sigmoid_hw(gbuf[j1]);
            const float f_ = sigmoid_hw(gbuf[HDIM + j1]);
            const float g_ = tanh_hw(gbuf[2 * HDIM + j1]);
            const float o_ = sigmoid_hw(gbuf[3 * HDIM + j1]);
            c1 = f_ * c1 + i_ * g_;
            const float h_ = o_ * tanh_hw(c1);
            vh[HDIM + j1] = h_;
            if (wave == 0) Hbuf[t * HDIM + j1] = h_;
            if (t + 1 < T_LEN)
                vh[j1] = sigmoid_hw(__builtin_fmaf(xs[t + 1], ws1, bs1));
        }
        if (has2) {
            const float i_ = sigmoid_hw(gbuf[j2]);
            const float f_ = sigmoid_hw(gbuf[HDIM + j2]);
            const float g_ = tanh_hw(gbuf[2 * HDIM + j2]);
            const float o_ = sigmoid_hw(gbuf[3 * HDIM + j2]);
            c2 = f_ * c2 + i_ * g_;
            const float h_ = o_ * tanh_hw(c2);
            vh[HDIM + j2] = h_;
            if (wave == 0) Hbuf[t * HDIM + j2] = h_;
            if (t + 1 < T_LEN)
                vh[j2] = sigmoid_hw(__builtin_fmaf(xs[t + 1], ws2, bs2));
        }
    }
}

__global__ void out_proj_kernel(const float* __restrict__ Hbuf,
                                const float* __restrict__ W_out,
                                const float* __restrict__ b_out,
                                float* __restrict__ out) {
    const int t = blockIdx.x * blockDim.x + threadIdx.x;
    if (t >= T_LEN) return;
    float s = b_out[0];
#pragma unroll
    for (int j = 0; j < HDIM; ++j) s += Hbuf[t * HDIM + j] * W_out[j];
    out[t] = s;
}

extern "C" void kernel_launch(void* const* d_in, const int* in_sizes, int n_in,
                              void* d_out, int out_size, void* d_ws, size_t ws_size,
                              hipStream_t stream) {
    const float* x     = (const float*)d_in[0];
    const float* W_sig = (const float*)d_in[1];
    const float* b_sig = (const float*)d_in[2];
    const float* W_ih  = (const float*)d_in[3];
    const float* b_ih  = (const float*)d_in[4];
    const float* W_hh  = (const float*)d_in[5];
    const float* b_hh  = (const float*)d_in[6];
    const float* W_out = (const float*)d_in[7];
    const float* b_out = (const float*)d_in[8];
    float* out  = (float*)d_out;
    float* Hbuf = (float*)d_ws;            // T*H floats = 1.6 MB scratch

    lstm_seq_kernel<<<1, NTHREADS, 0, stream>>>(x, W_sig, b_sig, W_ih, b_ih,
                                                W_hh, b_hh, Hbuf);
    out_proj_kernel<<<(T_LEN + 255) / 256, 256, 0, stream>>>(Hbuf, W_out, b_out, out);
}